// DetBenchEvalMultiScale_70231305224400
// MI455X (gfx1250) — compile-verified
//
#include <hip/hip_runtime.h>

// ---------------------------------------------------------------------------
// EfficientDet eval post-processing for MI455X (gfx1250, wave32).
// Pipeline: radix-histogram top-k (WMMA-accelerated chunk reduction) ->
// compact -> TDM-staged bitonic sort in LDS -> decode (exp/sigmoid) ->
// wave32 greedy NMS with TDM (tensor_load_to_lds) staging of boxes.
// ---------------------------------------------------------------------------

#define NCLS        90
#define TOPK        5000
#define CAP         8192          // candidate buffer per image (power of two)
#define BATCH       8
#define NDET        100

// workspace layout (bytes)
static constexpr size_t WS_HIST  = 0;                              // u32[8*65536]
static constexpr size_t WS_THR   = (size_t)BATCH * 65536 * 4;      // u32[8]
static constexpr size_t WS_CNT   = WS_THR + 256;                   // u32[8]
static constexpr size_t WS_CAND  = WS_CNT + 256;                   // u64[8*8192]
static constexpr size_t WS_BOX   = WS_CAND + (size_t)BATCH * CAP * 8;     // f32[8*5000*4]
static constexpr size_t WS_SCORE = WS_BOX + (size_t)BATCH * TOPK * 4 * 4; // f32[8*5000]
static constexpr size_t WS_CLS   = WS_SCORE + (size_t)BATCH * TOPK * 4;   // f32[8*5000]

typedef unsigned int       u32x4 __attribute__((ext_vector_type(4)));
typedef int                i32x4 __attribute__((ext_vector_type(4)));
typedef int                i32x8 __attribute__((ext_vector_type(8)));
typedef float              v2f   __attribute__((ext_vector_type(2)));
typedef float              v8f   __attribute__((ext_vector_type(8)));

#if __has_builtin(__builtin_amdgcn_tensor_load_to_lds) && __has_builtin(__builtin_amdgcn_s_wait_tensorcnt)
#define HAVE_TDM 1
#else
#define HAVE_TDM 0
#endif

#if __has_builtin(__builtin_amdgcn_wmma_f32_16x16x4_f32)
#define HAVE_WMMA4 1
#else
#define HAVE_WMMA4 0
#endif

// order-preserving float <-> uint key (descending float == descending key)
__device__ __forceinline__ unsigned int f2key(float f) {
    unsigned int u = __float_as_uint(f);
    return (u & 0x80000000u) ? ~u : (u | 0x80000000u);
}
__device__ __forceinline__ float key2f(unsigned int k) {
    unsigned int u = (k & 0x80000000u) ? (k & 0x7FFFFFFFu) : ~k;
    return __uint_as_float(u);
}

#if HAVE_TDM
// 1-row 2D TDM tile: nElem elements of (1<<dsLog2) bytes, contiguous
// global -> contiguous LDS. Issued once per executing wave; waits TENSORcnt.
__device__ __forceinline__ void tdm_load_1d(void* ldsDst, const void* gsrc,
                                            unsigned int nElem, unsigned int dsCode) {
    unsigned long long ga = (unsigned long long)gsrc;
    u32x4 g0;
    g0.x = 1u;                                                    // count=1 (user)
    g0.y = (unsigned int)(unsigned long long)ldsDst;              // lds_addr (low 32 of flat)
    g0.z = (unsigned int)ga;                                      // global_addr[31:0]
    g0.w = (unsigned int)((ga >> 32) & 0x01FFFFFFull) | 0x80000000u; // addr[56:32] | type=2
    i32x8 g1;
    g1[0] = (int)(dsCode << 16);                                  // data_size, wg_mask=0
    g1[1] = (int)((nElem & 0xFFFFu) << 16);                       // tensor_dim0[15:0]
    g1[2] = (int)((nElem >> 16) | (1u << 16));                    // tensor_dim0[31:16] | dim1=1
    g1[3] = (int)((nElem & 0xFFFFu) << 16);                       // tile_dim0 (< 65536)
    g1[4] = 1;                                                    // tile_dim1 = 1
    g1[5] = (int)nElem;                                           // tensor_dim0_stride
    g1[6] = 0; g1[7] = 0;
    i32x4 z4 = {0, 0, 0, 0};
#if defined(__clang_major__) && (__clang_major__ >= 23)
    i32x8 z8 = {0, 0, 0, 0, 0, 0, 0, 0};
    __builtin_amdgcn_tensor_load_to_lds(g0, g1, z4, z4, z8, 0);
#else
    __builtin_amdgcn_tensor_load_to_lds(g0, g1, z4, z4, 0);
#endif
    __builtin_amdgcn_s_wait_tensorcnt(0);
    asm volatile("" ::: "memory");
}
#endif

// ---------------------------------------------------------------------------
// Pass 1: 16-bit (high bits) histogram of keys, per image.
// ---------------------------------------------------------------------------
__global__ void k_hist(const float4* __restrict__ cls, int quarterPerImage,
                       unsigned int* __restrict__ hist) {
    int t = blockIdx.x * blockDim.x + threadIdx.x;
    if (t >= BATCH * quarterPerImage) return;
    int b = t / quarterPerImage;
    float4 v = cls[t];
    unsigned int* h = hist + (b << 16);
    atomicAdd(&h[f2key(v.x) >> 16], 1u);
    atomicAdd(&h[f2key(v.y) >> 16], 1u);
    atomicAdd(&h[f2key(v.z) >> 16], 1u);
    atomicAdd(&h[f2key(v.w) >> 16], 1u);
}

// ---------------------------------------------------------------------------
// Pass 2: find largest high-16 bin T with suffix-count(bins >= T) >= TOPK.
// Chunk sums (256 bins each) are reduced with V_WMMA_F32_16X16X4_F32:
// every lane pre-sums 64 bins into one exact f32 partial (all counts < 2^24),
// B holds 4 partials per column (lanes n, n+16 x 2 VGPRs), A = ones, so
// D[.][n] = exact chunk sum regardless of the K-slot assignment.
// ---------------------------------------------------------------------------
__global__ void k_select_thr(const unsigned int* __restrict__ hist,
                             unsigned int* __restrict__ thr) {
    __shared__ float chunk[256];
    int b = blockIdx.x;
    const unsigned int* h = hist + (b << 16);
    int wave = threadIdx.x >> 5;          // 0..7
    int lane = threadIdx.x & 31;
    int n    = lane & 15;                 // column = chunk within batch
    int hi   = lane >> 4;                 // 0: slots 0/1, 1: slots 2/3

    for (int batch = 0; batch < 2; ++batch) {
        int cbase   = wave * 32 + batch * 16;
        int chunkId = cbase + n;
        const unsigned int* cb = h + chunkId * 256;
        float p0 = 0.0f, p1 = 0.0f;
        int s0 = (hi * 2 + 0) * 64, s1 = (hi * 2 + 1) * 64;
        for (int i = 0; i < 64; ++i) {
            p0 += (float)cb[s0 + i];
            p1 += (float)cb[s1 + i];
        }
        float csum;
#if HAVE_WMMA4
        {
            v2f A = {1.0f, 1.0f};
            v2f B = {p0, p1};
            v8f C = {};
            C = __builtin_amdgcn_wmma_f32_16x16x4_f32(false, A, false, B,
                                                      (short)0, C, false, false);
            csum = C[0];                  // D[row 0][col n] = chunk sum
        }
#else
        {
            float s = p0 + p1;
            s += __shfl_xor(s, 16);
            csum = s;
        }
#endif
        if (lane < 16) chunk[chunkId] = csum;
    }
    __syncthreads();

    if (threadIdx.x == 0) {
        unsigned int cum = 0;
        int selChunk = 0;
        for (int c = 255; c >= 0; --c) {
            unsigned int cs = (unsigned int)chunk[c];
            if (cum + cs >= (unsigned)TOPK) { selChunk = c; break; }
            cum += cs;
        }
        int selBin = selChunk * 256;      // fallback
        for (int bin = selChunk * 256 + 255; bin >= selChunk * 256; --bin) {
            cum += h[bin];
            if (cum >= (unsigned)TOPK) { selBin = bin; break; }
        }
        thr[b] = (unsigned int)selBin;
    }
}

// ---------------------------------------------------------------------------
// Pass 3: compact survivors as (key<<32 | ~flatScoreIdx).
// flat = (anchorBase + pos*9 + a)*90 + c  matches jax reshape ordering.
// ---------------------------------------------------------------------------
__global__ void k_compact(const float* __restrict__ cls, int nPerImage, int HW,
                          int anchorBase, const unsigned int* __restrict__ thr,
                          unsigned int* __restrict__ cnt,
                          unsigned long long* __restrict__ cand) {
    int t = blockIdx.x * blockDim.x + threadIdx.x;
    if (t >= BATCH * nPerImage) return;
    int b = t / nPerImage;
    int e = t - b * nPerImage;                 // ch*HW + pos
    unsigned int k = f2key(cls[t]);
    if ((k >> 16) >= thr[b]) {
        int ch = e / HW, p = e - ch * HW;
        int a  = ch / NCLS, c = ch - a * NCLS; // channel = a*90 + class
        unsigned int flat = (unsigned int)(anchorBase + p * 9 + a) * 90u + (unsigned int)c;
        unsigned int pos = atomicAdd(&cnt[b], 1u);
        if (pos < CAP)
            cand[(size_t)b * CAP + pos] =
                ((unsigned long long)k << 32) | (unsigned int)(~flat);
    }
}

// ---------------------------------------------------------------------------
// Pass 4: TDM-stage 64 KB of candidates into LDS, bitonic sort (descending),
// decode top-5000 (gather box regressions + anchors, exp/sigmoid).
// ---------------------------------------------------------------------------
__global__ void k_sort_decode(const unsigned long long* __restrict__ cand,
                              const unsigned int* __restrict__ cnt,
                              const float* __restrict__ bx0, const float* __restrict__ bx1,
                              const float* __restrict__ bx2, const float* __restrict__ bx3,
                              const float* __restrict__ bx4,
                              const float* __restrict__ anchors,
                              float* __restrict__ boxes, float* __restrict__ scores,
                              float* __restrict__ clsf) {
    extern __shared__ unsigned long long sm[];
    int b = blockIdx.x;
    unsigned int n = cnt[b];
    if (n > CAP) n = CAP;

#if HAVE_TDM
    if (threadIdx.x < 32)   // wave 0 issues the DMA and drains TENSORcnt
        tdm_load_1d(sm, cand + (size_t)b * CAP, CAP, 3u /*8B*/);
    __syncthreads();
    for (int i = (int)n + threadIdx.x; i < CAP; i += blockDim.x)
        sm[i] = 0ull;       // zero the pad beyond the live candidates
#else
    for (int i = threadIdx.x; i < CAP; i += blockDim.x)
        sm[i] = (i < (int)n) ? cand[(size_t)b * CAP + i] : 0ull;
#endif
    __syncthreads();

    for (int k = 2; k <= CAP; k <<= 1) {
        for (int j = k >> 1; j > 0; j >>= 1) {
            for (int i = threadIdx.x; i < CAP; i += blockDim.x) {
                int l = i ^ j;
                if (l > i) {
                    unsigned long long a = sm[i], c = sm[l];
                    bool desc = ((i & k) == 0);
                    if (desc ? (a < c) : (a > c)) { sm[i] = c; sm[l] = a; }
                }
            }
            __syncthreads();
        }
    }

    for (int i = threadIdx.x; i < TOPK; i += blockDim.x) {
        unsigned long long sv = sm[i];
        unsigned int key  = (unsigned int)(sv >> 32);
        unsigned int flat = ~((unsigned int)sv);
        float logit = key2f(key);
        unsigned int aidx = flat / 90u;
        unsigned int c    = flat - aidx * 90u;

        int off, HW; const float* bp;
        if (aidx < 36864u)      { off = 0;     HW = 4096; bp = bx0; }
        else if (aidx < 46080u) { off = 36864; HW = 1024; bp = bx1; }
        else if (aidx < 48384u) { off = 46080; HW = 256;  bp = bx2; }
        else if (aidx < 48960u) { off = 48384; HW = 64;   bp = bx3; }
        else                    { off = 48960; HW = 16;   bp = bx4; }
        int rel = (int)aidx - off;
        int p = rel / 9, a = rel - p * 9;

        const float* base = bp + (size_t)b * 36 * HW + p;
        float ty = base[(a * 4 + 0) * HW];
        float tx = base[(a * 4 + 1) * HW];
        float th = base[(a * 4 + 2) * HW];
        float tw = base[(a * 4 + 3) * HW];

        float a0 = anchors[aidx * 4 + 0], a1 = anchors[aidx * 4 + 1];
        float a2 = anchors[aidx * 4 + 2], a3 = anchors[aidx * 4 + 3];
        float yca = (a0 + a2) * 0.5f, xca = (a1 + a3) * 0.5f;
        float ha = a2 - a0, wa = a3 - a1;
        float w  = expf(tw) * wa, h = expf(th) * ha;
        float yc = ty * ha + yca, xc = tx * wa + xca;

        float* ob = boxes + ((size_t)b * TOPK + i) * 4;
        ob[0] = xc - w * 0.5f;  ob[1] = yc - h * 0.5f;
        ob[2] = xc + w * 0.5f;  ob[3] = yc + h * 0.5f;
        scores[(size_t)b * TOPK + i] = 1.0f / (1.0f + expf(-logit));
        clsf[(size_t)b * TOPK + i]   = (float)(c + 1u);
    }
}

// ---------------------------------------------------------------------------
// Pass 5: greedy NMS, one wave32 per image. TDM stages the 5000x4 f32 box
// tile (80 KB) from global into LDS; wave waits on TENSORcnt then runs the
// serial scan with IoU checks spread across lanes (__any ballot).
// ---------------------------------------------------------------------------
__global__ void k_nms(const float* __restrict__ boxes, const float* __restrict__ scores,
                      const float* __restrict__ clsf, const float* __restrict__ img_scales,
                      float* __restrict__ out) {
    extern __shared__ float sb[];          // TOPK*4 floats (80 KB)
    __shared__ int keptIdx[NDET];
    int b = blockIdx.x;
    int lane = threadIdx.x;

#if HAVE_TDM
    tdm_load_1d(sb, boxes + (size_t)b * TOPK * 4, TOPK * 4, 2u /*4B*/);
#else
    for (int i = lane; i < TOPK * 4; i += 32)
        sb[i] = boxes[(size_t)b * TOPK * 4 + i];
#endif

    float scale = img_scales[b];
    int nkept = 0;
    for (int i = 0; i < TOPK && nkept < NDET; ++i) {
        float x1 = sb[i * 4 + 0], y1 = sb[i * 4 + 1];
        float x2 = sb[i * 4 + 2], y2 = sb[i * 4 + 3];
        float area_i = (x2 - x1) * (y2 - y1);
        int sup = 0;
        for (int t = lane; t < nkept; t += 32) {
            int kj = keptIdx[t];
            float kx1 = sb[kj * 4 + 0], ky1 = sb[kj * 4 + 1];
            float kx2 = sb[kj * 4 + 2], ky2 = sb[kj * 4 + 3];
            float iw = fminf(x2, kx2) - fmaxf(x1, kx1);
            float ih = fminf(y2, ky2) - fmaxf(y1, ky1);
            iw = fmaxf(iw, 0.0f); ih = fmaxf(ih, 0.0f);
            float inter = iw * ih;
            float ka = (kx2 - kx1) * (ky2 - ky1);
            float iou = inter / (area_i + ka - inter + 1e-8f);
            if (iou > 0.5f) { sup = 1; break; }
        }
        if (!__any(sup)) {
            if (lane == 0) {
                keptIdx[nkept] = i;
                float* o = out + ((size_t)b * NDET + nkept) * 6;
                o[0] = x1 * scale;        o[1] = y1 * scale;
                o[2] = (x2 - x1) * scale; o[3] = (y2 - y1) * scale;
                o[4] = scores[(size_t)b * TOPK + i];
                o[5] = clsf[(size_t)b * TOPK + i];
            }
            nkept++;
        }
    }
    for (int r = nkept; r < NDET; ++r)
        if (lane < 6) out[((size_t)b * NDET + r) * 6 + lane] = 0.0f;
}

// ---------------------------------------------------------------------------
extern "C" void kernel_launch(void* const* d_in, const int* in_sizes, int n_in,
                              void* d_out, int out_size, void* d_ws, size_t ws_size,
                              hipStream_t stream) {
    (void)in_sizes; (void)n_in; (void)out_size; (void)ws_size;
    // setup_inputs dict order: cls_p3, box_p3, cls_p4, box_p4, ..., image_scales, anchor_boxes
    const float* cls[5] = {(const float*)d_in[0], (const float*)d_in[2], (const float*)d_in[4],
                           (const float*)d_in[6], (const float*)d_in[8]};
    const float* box[5] = {(const float*)d_in[1], (const float*)d_in[3], (const float*)d_in[5],
                           (const float*)d_in[7], (const float*)d_in[9]};
    const float* scales  = (const float*)d_in[10];
    const float* anchors = (const float*)d_in[11];

    char* ws = (char*)d_ws;
    unsigned int*       hist   = (unsigned int*)(ws + WS_HIST);
    unsigned int*       thr    = (unsigned int*)(ws + WS_THR);
    unsigned int*       cnt    = (unsigned int*)(ws + WS_CNT);
    unsigned long long* cand   = (unsigned long long*)(ws + WS_CAND);
    float*              boxesd = (float*)(ws + WS_BOX);
    float*              scoresd= (float*)(ws + WS_SCORE);
    float*              clsfd  = (float*)(ws + WS_CLS);

    hipMemsetAsync(hist, 0, (size_t)BATCH * 65536 * 4, stream);
    hipMemsetAsync(cnt, 0, (size_t)BATCH * 4, stream);

    static const int HWs[5]  = {4096, 1024, 256, 64, 16};
    static const int AOFF[5] = {0, 36864, 46080, 48384, 48960};

    for (int l = 0; l < 5; ++l) {
        int nPer = 810 * HWs[l];
        int q = nPer / 4;
        int tot = BATCH * q;
        k_hist<<<(tot + 255) / 256, 256, 0, stream>>>((const float4*)cls[l], q, hist);
    }
    k_select_thr<<<BATCH, 256, 0, stream>>>(hist, thr);
    for (int l = 0; l < 5; ++l) {
        int nPer = 810 * HWs[l];
        int tot = BATCH * nPer;
        k_compact<<<(tot + 255) / 256, 256, 0, stream>>>(cls[l], nPer, HWs[l], AOFF[l],
                                                         thr, cnt, cand);
    }
    k_sort_decode<<<BATCH, 512, CAP * sizeof(unsigned long long), stream>>>(
        cand, cnt, box[0], box[1], box[2], box[3], box[4], anchors,
        boxesd, scoresd, clsfd);
    k_nms<<<BATCH, 32, (size_t)TOPK * 4 * sizeof(float), stream>>>(
        boxesd, scoresd, clsfd, scales, (float*)d_out);
}